// RegistrationRender_23433341567537
// MI455X (gfx1250) — compile-verified
//
#include <hip/hip_runtime.h>
#include <math.h>

typedef __attribute__((ext_vector_type(2))) float v2f;
typedef __attribute__((ext_vector_type(8))) float v8f;

#define NPTS        8192
#define NSRC        4096
#define NVIEW       6
#define NPIX        65536      // 256*256
#define PT_TILES    512        // 8192 pts / 16 per view
#define HALF_TILES  256        // src tiles (tgt is the next 256)
#define PIX_TILES   4096       // 65536 px / 16 per view
#define RHO2_HALF   5000.0f    // RHO^2 / 2

// Rotation matrices, row-major, exact float32 casts of numpy float64 values.
__constant__ float c_rot[NVIEW][9] = {
    { 1.0f, 0.0f, 0.0f,   0.0f, 1.0f, 0.0f,   0.0f, 0.0f, 1.0f },
    // Ry(pi/2): c=6.123234e-17, s=1
    { 6.123234e-17f, 0.0f, 1.0f,   0.0f, 1.0f, 0.0f,   -1.0f, 0.0f, 6.123234e-17f },
    // Ry(pi): c=-1, s=1.2246468e-16
    { -1.0f, 0.0f, 1.2246468e-16f,   0.0f, 1.0f, 0.0f,   -1.2246468e-16f, 0.0f, -1.0f },
    // Ry(3pi/2): c=-1.8369702e-16, s=-1
    { -1.8369702e-16f, 0.0f, -1.0f,   0.0f, 1.0f, 0.0f,   1.0f, 0.0f, -1.8369702e-16f },
    // Rx(pi/2): c=6.123234e-17, s=1
    { 1.0f, 0.0f, 0.0f,   0.0f, 6.123234e-17f, -1.0f,   0.0f, 1.0f, 6.123234e-17f },
    // Rx(3pi/2): c=-1.8369702e-16, s=-1
    { 1.0f, 0.0f, 0.0f,   0.0f, -1.8369702e-16f, 1.0f,   0.0f, -1.0f, -1.8369702e-16f },
};

// ---------------------------------------------------------------------------
// Pass 1: per-view min/max of rotated depth (z) over all 8192 points.
// ---------------------------------------------------------------------------
__global__ void minmax_kernel(const float* __restrict__ src,
                              const float* __restrict__ tgt,
                              float* __restrict__ mm) {
    __shared__ float smin[256];
    __shared__ float smax[256];
    const int v   = blockIdx.x;
    const int tid = threadIdx.x;
    const float r20 = c_rot[v][6], r21 = c_rot[v][7], r22 = c_rot[v][8];
    float zmin = 3.0e38f, zmax = -3.0e38f;
    for (int p = tid; p < NPTS; p += 256) {
        const float* pt = (p < NSRC) ? (src + 3 * p) : (tgt + 3 * (p - NSRC));
        float z = r20 * pt[0] + r21 * pt[1] + r22 * pt[2];
        zmin = fminf(zmin, z);
        zmax = fmaxf(zmax, z);
    }
    smin[tid] = zmin; smax[tid] = zmax;
    __syncthreads();
    for (int s = 128; s > 0; s >>= 1) {
        if (tid < s) {
            smin[tid] = fminf(smin[tid], smin[tid + s]);
            smax[tid] = fmaxf(smax[tid], smax[tid + s]);
        }
        __syncthreads();
    }
    if (tid == 0) { mm[2 * v] = smin[0]; mm[2 * v + 1] = smax[0]; }
}

// ---------------------------------------------------------------------------
// Pass 2: build packed WMMA A-tiles.
// A-matrix (16x4 f32) lane layout: lanes 0-15 hold (K0,K1) of row M=lane,
// lanes 16-31 hold (K2,K3) of row M=lane-16. Per tile: 32 x float2 = 256 B.
// Row i: K0 = 1e4*x0, K1 = 1e4*x1, K2 = a_i = log(pv_i) - 5000*(x0^2+x1^2), K3 = 0.
// ---------------------------------------------------------------------------
__global__ void apack_kernel(const float* __restrict__ src,
                             const float* __restrict__ tgt,
                             const float* __restrict__ mm,
                             float2* __restrict__ apack) {
    const int idx = blockIdx.x * 256 + threadIdx.x;   // [0, 6*8192)
    const int v = idx >> 13;
    const int p = idx & (NPTS - 1);
    const float* pt = (p < NSRC) ? (src + 3 * p) : (tgt + 3 * (p - NSRC));
    const float px = pt[0], py = pt[1], pz = pt[2];
    const float x0 = c_rot[v][0] * px + c_rot[v][1] * py + c_rot[v][2] * pz;
    const float x1 = c_rot[v][3] * px + c_rot[v][4] * py + c_rot[v][5] * pz;
    const float z  = c_rot[v][6] * px + c_rot[v][7] * py + c_rot[v][8] * pz;
    const float dmin = mm[2 * v], dmax = mm[2 * v + 1];
    const float pv = 1.0f - (z - dmin) / (dmax - dmin);
    float lg = (pv > 0.0f) ? logf(pv) : -3.0e38f;
    float a  = lg - RHO2_HALF * (x0 * x0 + x1 * x1);
    if (!(a > -3.0e38f)) a = -3.0e38f;   // avoid -inf entering WMMA
    const int tile = p >> 4;
    const int r    = p & 15;
    float2* base = apack + (size_t)(v * PT_TILES + tile) * 32;
    base[r]      = make_float2(1.0e4f * x0, 1.0e4f * x1);
    base[16 + r] = make_float2(a, 0.0f);
}

// ---------------------------------------------------------------------------
// Pass 3: render. One wave per 16-pixel tile; 256 WMMA + max iterations per
// src/tgt half. B-col j: (p0, p1, 1, 0) -> lanes 0-15 hold (p0,p1),
// lanes 16-31 hold (1,0). D_ij = a_i + 1e4*x_i . p_j = log-space score.
// ---------------------------------------------------------------------------
__global__ void render_kernel(const float2* __restrict__ apack,
                              float* __restrict__ out) {
    const int lane  = threadIdx.x & 31;
    const int wv    = threadIdx.x >> 5;
    const int ptile = blockIdx.x * 8 + wv;          // [0, 6*4096)
    const int v     = ptile >> 12;
    const int tile  = ptile & (PIX_TILES - 1);
    const int j     = tile * 16 + (lane & 15);      // pixel id for this lane slot
    const float p0  = ((float)(j & 255) - 127.5f) * (1.0f / 128.0f);
    const float p1  = ((float)(j >> 8)  - 127.5f) * (1.0f / 128.0f);

    v2f b;
    if (lane < 16) { b[0] = p0;   b[1] = p1;   }
    else           { b[0] = 1.0f; b[1] = 0.0f; }

    const float2* Abase = apack + (size_t)v * (PT_TILES * 32) + lane;
    const float pixterm = -RHO2_HALF * (p0 * p0 + p1 * p1);

    for (int half = 0; half < 2; ++half) {          // 0 = src image, 1 = tgt image
        const float2* A = Abase + (size_t)half * (HALF_TILES * 32);
        v8f macc;
#pragma unroll
        for (int r = 0; r < 8; ++r) macc[r] = -3.0e38f;

#pragma unroll 4
        for (int t = 0; t < HALF_TILES; ++t) {
            float2 av = A[(size_t)t * 32];          // one global_load_b64 / lane
            v2f a; a[0] = av.x; a[1] = av.y;
            v8f c = {};
            v8f d = __builtin_amdgcn_wmma_f32_16x16x4_f32(
                false, a, false, b, (short)0, c, false, false);
#pragma unroll
            for (int r = 0; r < 8; ++r) macc[r] = fmaxf(macc[r], d[r]);
        }

        // rows 0-7 / 8-15 live in the two lane halves: reduce 8 VGPRs, then SWAPX16.
        float m = fmaxf(fmaxf(fmaxf(macc[0], macc[1]), fmaxf(macc[2], macc[3])),
                        fmaxf(fmaxf(macc[4], macc[5]), fmaxf(macc[6], macc[7])));
        m = fmaxf(m, __int_as_float(
                         __builtin_amdgcn_ds_swizzle(__float_as_int(m), 0x401F)));

        if (lane < 16) {
            float val = 2.0f * expf(m + pixterm) - 1.0f;   // (v - 0.5)/0.5
            out[(size_t)(v * 2 + half) * NPIX + j] = val;
        }
    }
}

// ---------------------------------------------------------------------------
extern "C" void kernel_launch(void* const* d_in, const int* in_sizes, int n_in,
                              void* d_out, int out_size, void* d_ws, size_t ws_size,
                              hipStream_t stream) {
    (void)in_sizes; (void)n_in; (void)out_size; (void)ws_size;
    const float* src = (const float*)d_in[0];
    const float* tgt = (const float*)d_in[1];
    float* out = (float*)d_out;

    float2* apack = (float2*)d_ws;                                  // 786432 B
    float*  mm    = (float*)((char*)d_ws +
                             (size_t)NVIEW * PT_TILES * 32 * sizeof(float2));

    minmax_kernel<<<NVIEW, 256, 0, stream>>>(src, tgt, mm);
    apack_kernel<<<(NVIEW * NPTS) / 256, 256, 0, stream>>>(src, tgt, mm, apack);
    render_kernel<<<(NVIEW * PIX_TILES) / 8, 256, 0, stream>>>(apack, out);
}